// Pinn_86964497809860
// MI455X (gfx1250) — compile-verified
//
#include <hip/hip_runtime.h>
#include <hip/hip_bf16.h>

// PINN jet-propagation kernel for MI455X (gfx1250).
// Jet = [val, d/dq0..2, 6 second-order, 10 third-order] = 20 channels/point.
// All linear layers -> fp32 WMMA (V_WMMA_F32_16X16X4_F32); tanh -> Faa di Bruno.

#define HIDDEN 128
#define BLOCKS 8
#define NPTS 4096
#define PTS_PER_WG 4
#define CH 20
#define COLS (PTS_PER_WG * CH)   // 80
#define CT_TILES (COLS / 16)     // 5

typedef float v2f __attribute__((ext_vector_type(2)));
typedef float v8f __attribute__((ext_vector_type(8)));

struct PinnArgs {
  const float *x, *y, *t, *u, *v;
  const float *fw, *fb, *lw, *lb;
  const float *w1[BLOCKS], *b1[BLOCKS], *w2[BLOCKS], *b2[BLOCKS];
  const float *lam1, *lam2, *minx, *maxx;
  float *out, *ws;
};

__device__ __forceinline__ v8f wmma_f32(v2f a, v2f b, v8f c) {
#if defined(__gfx1250__) && __has_builtin(__builtin_amdgcn_wmma_f32_16x16x4_f32)
  // 8 args: (neg_a, A, neg_b, B, c_mod, C, reuse_a, reuse_b)
  return __builtin_amdgcn_wmma_f32_16x16x4_f32(
      false, a, false, b, (short)0, c, false, false);
#else
  c[0] += a[0] * b[0];  // host-pass placeholder; never used on device
  return c;
#endif
}

__global__ void __launch_bounds__(256, 2)
pinn_kernel(PinnArgs args) {
  extern __shared__ float smem[];
  float* Hs = smem;                    // 128 x 80 residual-stream jet
  float* As = smem + HIDDEN * COLS;    // 128 x 80 activation chunk / scratch

  const int tid   = threadIdx.x;
  const int lane  = tid & 31;
  const int w     = tid >> 5;          // wave id, 0..7 -> row tile
  const int m     = lane & 15;         // M (A) or N (B/C) index within tile
  const int kb    = (lane >> 4) << 1;  // K sub-offset: 0 (lanes 0-15) / 2 (16-31)
  const int rhalf = (lane >> 4) << 3;  // C/D row half: 0 / 8
  const int gbase = blockIdx.x * PTS_PER_WG;

  // ------------ seed input jet: h = first_w @ z + first_b ------------
  for (int pr = tid; pr < HIDDEN * PTS_PER_WG; pr += blockDim.x) {
    const int row = pr >> 2;
    const int p   = pr & 3;
    const float q[3] = { args.x[gbase + p], args.y[gbase + p], args.t[gbase + p] };
    float* Hc = Hs + row * COLS + p * CH;
    float v0 = args.fb[row];
#pragma unroll
    for (int j = 0; j < 3; ++j) {
      const float inv = 2.0f / (args.maxx[j] - args.minx[j]);
      const float z   = (q[j] - args.minx[j]) * inv - 1.0f;
      const float fwj = args.fw[row * 3 + j];
      v0 += fwj * z;
      Hc[1 + j] = fwj * inv;           // d h/d q_j  (chain through scaling)
    }
    Hc[0] = v0;
#pragma unroll
    for (int c = 4; c < CH; ++c) Hc[c] = 0.0f;
  }
  __syncthreads();

  // ------------ residual blocks ------------
#pragma unroll 1
  for (int blk = 0; blk < BLOCKS; ++blk) {
    const float* W1 = args.w1[blk];
    const float* B1 = args.b1[blk];
    const float* W2 = args.w2[blk];
    const float* B2 = args.b2[blk];
    __builtin_prefetch(W1 + (tid << 4), 0, 1);   // global_prefetch_b8
    __builtin_prefetch(W2 + (tid << 4), 0, 1);

    v8f acc2[CT_TILES];
    const v8f vzero = {};
#pragma unroll
    for (int ct = 0; ct < CT_TILES; ++ct) acc2[ct] = vzero;

#pragma unroll 1
    for (int chunk = 0; chunk < 4; ++chunk) {
      // ---- GEMM1: S[16x80] = W1[rowtile, 0:128] @ H[128x80] ----
      v2f aw[32];
      const float* w1row = W1 + (chunk * HIDDEN + w * 16 + m) * HIDDEN;
#pragma unroll
      for (int kk = 0; kk < 32; ++kk)
        aw[kk] = *(const v2f*)(w1row + kk * 4 + kb);

#pragma unroll
      for (int ct = 0; ct < CT_TILES; ++ct) {
        const int col = ct * 16 + m;
        v8f acc = vzero;
#pragma unroll
        for (int kk = 0; kk < 32; ++kk) {
          v2f b;
          b.x = Hs[(kk * 4 + kb) * COLS + col];
          b.y = Hs[(kk * 4 + kb + 1) * COLS + col];
          acc = wmma_f32(aw[kk], b, acc);
        }
#pragma unroll
        for (int r = 0; r < 8; ++r)
          As[(w * 16 + rhalf + r) * COLS + col] = acc[r];
      }
      __syncthreads();

      // ---- tanh jet (Faa di Bruno, order 3) in place on As ----
      for (int pr = tid; pr < HIDDEN * PTS_PER_WG; pr += blockDim.x) {
        const int row = pr >> 2;
        const int p   = pr & 3;
        float* S = As + row * COLS + p * CH;
        float s[20];
#pragma unroll
        for (int c = 0; c < CH; ++c) s[c] = S[c];
        s[0] += B1[chunk * HIDDEN + row];
        const float a0 = tanhf(s[0]);
        const float f1 = 1.0f - a0 * a0;
        const float f2 = -2.0f * a0 * f1;
        const float f3 = f1 * (6.0f * a0 * a0 - 2.0f);
        float o[20];
        o[0] = a0;
        o[1] = f1 * s[1]; o[2] = f1 * s[2]; o[3] = f1 * s[3];
        const int PI[6] = {0,0,0,1,1,2}, PJ[6] = {0,1,2,1,2,2};
#pragma unroll
        for (int n2 = 0; n2 < 6; ++n2)
          o[4 + n2] = f1 * s[4 + n2] + f2 * s[1 + PI[n2]] * s[1 + PJ[n2]];
        const int I2[3][3] = {{4,5,6},{5,7,8},{6,8,9}};
        const int TI[10] = {0,0,0,0,0,0,1,1,1,2};
        const int TJ[10] = {0,0,0,1,1,2,1,1,2,2};
        const int TK[10] = {0,1,2,1,2,2,1,2,2,2};
#pragma unroll
        for (int n3 = 0; n3 < 10; ++n3) {
          const int i = TI[n3], j = TJ[n3], k = TK[n3];
          const float term = s[1 + i] * s[I2[j][k]] +
                             s[1 + j] * s[I2[i][k]] +
                             s[1 + k] * s[I2[i][j]];
          const float cube = s[1 + i] * s[1 + j] * s[1 + k];
          o[10 + n3] = f1 * s[10 + n3] + f2 * term + f3 * cube;
        }
#pragma unroll
        for (int c = 0; c < CH; ++c) S[c] = o[c];
      }
      __syncthreads();

      // ---- GEMM2: Hacc[16x80] += W2[rowtile, chunk] @ A[128x80] ----
      const float* w2row = W2 + (w * 16 + m) * (4 * HIDDEN) + chunk * HIDDEN;
#pragma unroll
      for (int kk = 0; kk < 32; ++kk)
        aw[kk] = *(const v2f*)(w2row + kk * 4 + kb);
#pragma unroll
      for (int ct = 0; ct < CT_TILES; ++ct) {
        const int col = ct * 16 + m;
#pragma unroll
        for (int kk = 0; kk < 32; ++kk) {
          v2f b;
          b.x = As[(kk * 4 + kb) * COLS + col];
          b.y = As[(kk * 4 + kb + 1) * COLS + col];
          acc2[ct] = wmma_f32(aw[kk], b, acc2[ct]);
        }
      }
      __syncthreads();
    }

    // ---- residual write-back: H += fc2(acc) + fc2_b (value channel) ----
#pragma unroll
    for (int ct = 0; ct < CT_TILES; ++ct) {
      const int col = ct * 16 + m;
#pragma unroll
      for (int r = 0; r < 8; ++r) {
        const int row = w * 16 + rhalf + r;
        float add = acc2[ct][r];
        if ((col % CH) == 0) add += B2[row];
        Hs[row * COLS + col] += add;
      }
    }
    __syncthreads();
  }

  // ------------ output heads: psi/p jets = last_w @ h (+ last_b) ------------
  if (tid < PTS_PER_WG * 2 * CH) {
    const int p = tid / (2 * CH);
    const int rem = tid % (2 * CH);
    const int o = rem / CH;
    const int c = rem % CH;
    float sum = (c == 0) ? args.lb[o] : 0.0f;
    for (int r = 0; r < HIDDEN; ++r)
      sum += args.lw[o * HIDDEN + r] * Hs[r * COLS + p * CH + c];
    As[p * (2 * CH) + o * CH + c] = sum;
  }
  __syncthreads();

  // ------------ physics residuals per point ------------
  if (tid < PTS_PER_WG) {
    const int p  = tid;
    const int gi = gbase + p;
    const float* J = As + p * (2 * CH);  // [0..19]=psi jet, [20..39]=p jet
    const float l1 = args.lam1[0], l2 = args.lam2[0];
    const float u_pred =  J[2];          // psi_y
    const float v_pred = -J[1];          // -psi_x
    const float u_t =  J[8],  u_x =  J[5],  u_y =  J[7];
    const float v_t = -J[6],  v_x = -J[4],  v_y = -J[5];
    const float u_xx =  J[11], u_yy =  J[16];
    const float v_xx = -J[10], v_yy = -J[13];
    const float p_pred = J[20], p_x = J[21], p_y = J[22];
    const float f_u = l1 * (u_t + u_pred * u_x + v_pred * u_y) + p_x - l2 * (u_xx + u_yy);
    const float f_v = l1 * (v_t + u_pred * v_x + v_pred * v_y) + p_y - l2 * (v_xx + v_yy);
    args.out[gi * 3 + 0] = p_pred;
    args.out[gi * 3 + 1] = u_pred;
    args.out[gi * 3 + 2] = v_pred;
    args.out[NPTS * 3 + 1 + gi] = f_u;
    args.out[NPTS * 3 + 1 + NPTS + gi] = f_v;
    const float du = u_pred - args.u[gi];
    const float dv = v_pred - args.v[gi];
    atomicAdd(&args.ws[0], du * du);
    atomicAdd(&args.ws[1], dv * dv);
    atomicAdd(&args.ws[2], f_u * f_u);
    atomicAdd(&args.ws[3], f_v * f_v);
  }
}

__global__ void pinn_loss_combine(const float* ws, float* out) {
  if (threadIdx.x == 0 && blockIdx.x == 0) {
    const float invN = 1.0f / (float)NPTS;
    const float t0 = ws[0] * invN;
    const float t1 = ws[1] * invN;
    const float t2 = 0.1f * ws[2] * invN;
    const float t3 = 0.1f * ws[3] * invN;
    out[NPTS * 3] = 0.25f * (t0 + t1 + t2 + t3);
  }
}

extern "C" void kernel_launch(void* const* d_in, const int* in_sizes, int n_in,
                              void* d_out, int out_size, void* d_ws, size_t ws_size,
                              hipStream_t stream) {
  (void)in_sizes; (void)n_in; (void)out_size; (void)ws_size;
  PinnArgs a;
  a.x = (const float*)d_in[0];
  a.y = (const float*)d_in[1];
  a.t = (const float*)d_in[2];
  a.u = (const float*)d_in[3];
  a.v = (const float*)d_in[4];
  a.fw = (const float*)d_in[5];
  a.fb = (const float*)d_in[6];
  a.lw = (const float*)d_in[7];
  a.lb = (const float*)d_in[8];
  for (int b = 0; b < BLOCKS; ++b) {
    a.w1[b] = (const float*)d_in[9 + 4 * b];
    a.b1[b] = (const float*)d_in[10 + 4 * b];
    a.w2[b] = (const float*)d_in[11 + 4 * b];
    a.b2[b] = (const float*)d_in[12 + 4 * b];
  }
  a.lam1 = (const float*)d_in[41];
  a.lam2 = (const float*)d_in[42];
  a.minx = (const float*)d_in[43];
  a.maxx = (const float*)d_in[44];
  a.out = (float*)d_out;
  a.ws  = (float*)d_ws;

  hipMemsetAsync(d_ws, 0, 4 * sizeof(float), stream);
  const size_t smem = 2u * HIDDEN * COLS * sizeof(float);  // 80 KB
  pinn_kernel<<<NPTS / PTS_PER_WG, 256, smem, stream>>>(a);
  pinn_loss_combine<<<1, 32, 0, stream>>>((const float*)d_ws, (float*)d_out);
}